// GCN_20753281975108
// MI455X (gfx1250) — compile-verified
//
#include <hip/hip_runtime.h>

typedef __attribute__((ext_vector_type(2))) float v2f;
typedef __attribute__((ext_vector_type(8))) float v8f;

#define NN    50000
#define MM    32
#define DD    128
#define KK    8
#define BIGF  1e11f
#define ROWF  136                 // padded LDS row stride (floats): 128 + 8
#define ROWB  (ROWF * 4)          // 544 bytes; (2*136)%64 = 16 -> halves on disjoint banks

// 32 async b128 row copies; global stride (512 B) rides the shared `offset:`
// immediate, the LDS register only carries the 32 B/row padding residue:
//   lds = (base + 32*m) + 512*m,  global = ga + 512*m
template <int M>
__device__ __forceinline__ void stage_rows(unsigned la, unsigned long long ga) {
    if constexpr (M < MM) {
        asm volatile("global_load_async_to_lds_b128 %0, %1, off offset:%c2"
                     :: "v"(la), "v"(ga), "n"(M * DD * 4) : "memory");
        stage_rows<M + 1>(la + (ROWB - DD * 4), ga);
    }
}

__global__ __launch_bounds__(256) void gcn_topk_wmma(
    const float*     __restrict__ interact,   // [N, 32]
    const float*     __restrict__ initial,    // [N, 32]
    const float*     __restrict__ src,        // [N, 32, 128]
    const float*     __restrict__ buf,        // [N, 128]
    const float*     __restrict__ keep_rate_p,// [1]
    const long long* __restrict__ source_id,  // [N, 32] int64
    float*           __restrict__ out_agg,    // [N, 128]
    long long*       __restrict__ out_ids)    // [N, 8]
{
    __shared__ __align__(16) float tile[8][MM * ROWF];   // 136 KB: staged src tiles
    __shared__ __align__(16) float sbuf[8][MM];          // per-wave scores
    __shared__ __align__(16) float abuf[8][MM];          // per-wave adj row

    const int lane = threadIdx.x & 31;          // mailbox slot m
    const int wave = threadIdx.x >> 5;
    const int n    = blockIdx.x * 8 + wave;     // grid sized so n < N always (EXEC all-1s)

    // ---- async stream of this node's 16 KB tile: HBM b128 -> LDS ----
    stage_rows<0>((unsigned)(uintptr_t)(&tile[wave][0]) + ((unsigned)lane << 4),
                  (unsigned long long)(uintptr_t)(src + (size_t)n * (MM * DD))
                      + ((unsigned)lane << 4));

    const float keep = keep_rate_p[0];

    // ---- per-lane mailbox entry; publish scores to LDS for broadcast reads ----
    const float s    = interact[(size_t)n * MM + lane];
    const float init = initial [(size_t)n * MM + lane];
    sbuf[wave][lane] = s;

    // ---- rank in descending order (ties -> lower index, like lax.top_k) ----
    // All 32 scores via 8 broadcast ds_load_b128 (uniform address -> no conflicts).
    int rank = 0;
    {
        const float4* sv = (const float4*)&sbuf[wave][0];
        #pragma unroll
        for (int q = 0; q < 8; ++q) {
            float4 v = sv[q];
            const int j = 4 * q;
            rank += (v.x > s) || (v.x == s && (j + 0) < lane);
            rank += (v.y > s) || (v.y == s && (j + 1) < lane);
            rank += (v.z > s) || (v.z == s && (j + 2) < lane);
            rank += (v.w > s) || (v.w == s && (j + 3) < lane);
        }
    }
    const bool topk = rank < KK;

    // ---- second output: gathered source ids in top-k order ----
    if (topk) out_ids[(size_t)n * KK + rank] = source_id[(size_t)n * MM + lane];

    // ---- masked sigmoid blended with initial scores ----
    float x   = topk ? s : -BIGF;
    float sig = 1.0f / (1.0f + expf(-x));       // expf(1e11)=inf -> 0, matches ref
    float adj = (1.0f - keep) * sig + keep * init;
    abuf[wave][lane] = adj;

    // ---- B operands straight from LDS: adj[4c+half2], adj[4c+half2+1] are
    // adjacent and 8B-aligned -> broadcast ds_load_2addr_b64 pairs.
    // Slot convention matches the documented 16x4 f32 A layout (VGPR0 holds
    // k={0,2} by lane-half, VGPR1 k={1,3}); A uses the same convention, so the
    // k pairing is consistent under any hardware slot order.
    const int half2 = (lane >> 4) << 1;         // 0 or 2
    v2f bv[8];
    #pragma unroll
    for (int cc = 0; cc < 8; ++cc)
        bv[cc] = *(const v2f*)&abuf[wave][4 * cc + half2];

    // ---- coefficient = 1 + sum_m adj: halves each hold a disjoint partial ----
    float partial = 0.0f;
    #pragma unroll
    for (int cc = 0; cc < 8; ++cc) partial += bv[cc].x + bv[cc].y;
    const float total    = partial + __shfl_xor(partial, 16, 32);  // single exchange
    const float inv_coef = 1.0f / (total + 1.0f);

    // ---- wait for the staged tile (per-wave private buffer: no barrier) ----
    asm volatile("s_wait_asynccnt 0x0" ::: "memory");

    // ---- weighted neighbor reduction via V_WMMA_F32_16X16X4_F32 ----
    // A (16x4): row i = d-index (16t+i), k = m within chunk, from LDS:
    //   lane L<16 : vgpr0 = tile[4c+0][16t+L],    vgpr1 = tile[4c+1][16t+L]
    //   lane L>=16: vgpr0 = tile[4c+2][16t+L-16], vgpr1 = tile[4c+3][...]
    // B: adj[4c+k] broadcast across all 16 columns -> every column of D holds
    // the true reduction, so any lane can commit results.
    const float* sA = &tile[wave][(size_t)half2 * ROWF + (lane & 15)];

    #pragma unroll
    for (int t = 0; t < 8; ++t) {               // d-tile: d in [16t, 16t+16)
        v8f acc = {};
        #pragma unroll
        for (int cc = 0; cc < 8; ++cc) {        // m-chunk: m in [4c, 4c+4)
            v2f a;
            a.x = sA[(4 * cc + 0) * ROWF + 16 * t];   // fuses to ds_load_2addr_b32
            a.y = sA[(4 * cc + 1) * ROWF + 16 * t];
            // (neg_a, A, neg_b, B, c_mod, C, reuse_a, reuse_b)
            acc = __builtin_amdgcn_wmma_f32_16x16x4_f32(
                false, a, false, bv[cc], (short)0, acc, false, false);
        }
        // acc[r]: lanes 0-15 hold out[16t+r], lanes 16-31 hold out[16t+r+8];
        // columns are replicas, so lanes 0 and 16 commit 8 floats each (2xb128).
        if ((lane & 15) == 0) {
            const int dbase = 16 * t + ((lane >> 4) << 3);
            const float4* bp = (const float4*)(buf     + (size_t)n * DD + dbase);
            float4*       op = (float4*)      (out_agg + (size_t)n * DD + dbase);
            float4 r0, r1;
            r0.x = (acc[0] + bp[0].x) * inv_coef;
            r0.y = (acc[1] + bp[0].y) * inv_coef;
            r0.z = (acc[2] + bp[0].z) * inv_coef;
            r0.w = (acc[3] + bp[0].w) * inv_coef;
            r1.x = (acc[4] + bp[1].x) * inv_coef;
            r1.y = (acc[5] + bp[1].y) * inv_coef;
            r1.z = (acc[6] + bp[1].z) * inv_coef;
            r1.w = (acc[7] + bp[1].w) * inv_coef;
            op[0] = r0;
            op[1] = r1;
        }
    }
}

extern "C" void kernel_launch(void* const* d_in, const int* in_sizes, int n_in,
                              void* d_out, int out_size, void* d_ws, size_t ws_size,
                              hipStream_t stream) {
    const float*     interact = (const float*)d_in[0];
    const float*     initial  = (const float*)d_in[1];
    const float*     src      = (const float*)d_in[2];
    const float*     buf      = (const float*)d_in[3];
    // d_in[4] = W : dead code in the reference (relu(fc(.)) result discarded)
    const float*     keep     = (const float*)d_in[5];
    const long long* sid      = (const long long*)d_in[6];

    float*     out_agg = (float*)d_out;                           // N*128 f32
    long long* out_ids = (long long*)(out_agg + (size_t)NN * DD); // N*8 i64

    dim3 grid(NN / 8);   // 6250 blocks x 8 waves/block = 50000 nodes exactly
    dim3 block(256);
    gcn_topk_wmma<<<grid, block, 0, stream>>>(interact, initial, src, buf,
                                              keep, sid, out_agg, out_ids);
}